// GraphConv_20289425506353
// MI455X (gfx1250) — compile-verified
//
#include <hip/hip_runtime.h>

typedef float v2f __attribute__((ext_vector_type(2)));
typedef float v8f __attribute__((ext_vector_type(8)));

#define C64 64

// ---------------------------------------------------------------------------
// Kernel 1: fill scatter-max workspace with -inf (float4 vectorized)
// ---------------------------------------------------------------------------
__global__ void k_init(float4* __restrict__ p, int n4) {
    int g = blockIdx.x * blockDim.x + threadIdx.x;
    if (g < n4) {
        float ninf = -__builtin_inff();
        p[g] = make_float4(ninf, ninf, ninf, ninf);
    }
}

// ---------------------------------------------------------------------------
// Kernel 2: scatter-max of x[src] into sm[dst] (per channel).
// Uses the order-preserving integer reinterpretation of IEEE floats:
//   non-negative floats ordered as int, negative floats reverse-ordered as uint.
// Lowers to native global_atomic_max_i32 / global_atomic_min_u32 (no CAS loop).
// ---------------------------------------------------------------------------
__device__ __forceinline__ void atomic_max_f32(float* addr, float val) {
    if (!__builtin_signbitf(val)) {
        atomicMax((int*)addr, __float_as_int(val));
    } else {
        atomicMin((unsigned int*)addr, __float_as_uint(val));
    }
}

__global__ void k_edge(const float* __restrict__ x,
                       const int* __restrict__ src,
                       const int* __restrict__ dst,
                       float* __restrict__ sm, int E) {
    // one thread handles 4 channels of one edge -> 16 threads per edge
    int g = blockIdx.x * blockDim.x + threadIdx.x;
    int e = g >> 4;
    if (e >= E) return;
    int c = (g & 15) * 4;
    int s = src[e];
    int d = dst[e];
    const float4 v = *(const float4*)(x + (size_t)s * C64 + c);
    float* p = sm + (size_t)d * C64 + c;
    atomic_max_f32(p + 0, v.x);
    atomic_max_f32(p + 1, v.y);
    atomic_max_f32(p + 2, v.z);
    atomic_max_f32(p + 3, v.w);
}

// ---------------------------------------------------------------------------
// Kernel 3: out = relu([x | (sm - x, -inf->0)] @ W + b) via V_WMMA_F32_16X16X4_F32
// Block = 256 threads = 8 waves; block owns 32 rows; each wave owns a 16x16
// output tile (2 row-tiles x 4 col-tiles). K = 128 consumed 4 at a time.
// ---------------------------------------------------------------------------
__global__ __launch_bounds__(256) void k_gemm(const float* __restrict__ x,
                                              const float* __restrict__ sm,
                                              const float* __restrict__ W,   // [128,64] row-major
                                              const float* __restrict__ bias,// [64]
                                              float* __restrict__ out,       // [N,64]
                                              int N) {
    __shared__ float hs[32][132];   // 32x128 h tile, padded stride 132 (bank-conflict break)

    const int row0 = blockIdx.x * 32;
    const int t = threadIdx.x;

    // ---- stage h = [x | x_j] into LDS, fusing x_j = (sm==-inf ? 0 : sm - x)
    #pragma unroll
    for (int i = 0; i < 16; ++i) {
        int idx = t + i * 256;          // 0..4095
        int r = idx >> 7;               // 0..31
        int c = idx & 127;              // 0..127
        float v;
        if (c < C64) {
            v = x[(size_t)(row0 + r) * C64 + c];
        } else {
            int cc = c - C64;
            float m = sm[(size_t)(row0 + r) * C64 + cc];
            if (m == -__builtin_inff()) {
                v = 0.0f;
            } else {
                v = m - x[(size_t)(row0 + r) * C64 + cc];
            }
        }
        hs[r][c] = v;
    }
    __syncthreads();

    const int lane = t & 31;
    const int wave = t >> 5;
    const int r0   = (wave >> 2) * 16;  // row-tile base within block: 0 or 16
    const int c0   = (wave & 3)  * 16;  // col-tile base: 0,16,32,48
    const int half = lane >> 4;         // 0: lanes 0-15, 1: lanes 16-31
    const int lm   = lane & 15;

    v8f acc = {0.f, 0.f, 0.f, 0.f, 0.f, 0.f, 0.f, 0.f};

    #pragma unroll 4
    for (int k0 = 0; k0 < 128; k0 += 4) {
        const int ka = k0 + half * 2;   // this half-wave's K pair

        // A fragment: 16x4 f32, lane = M (=lm), VGPR0/1 = K = ka, ka+1  (LDS)
        v2f a;
        a.x = hs[r0 + lm][ka];
        a.y = hs[r0 + lm][ka + 1];

        // B fragment: 4x16 f32, lane = N (=c0+lm), VGPR0/1 = K = ka, ka+1 (global, L0-resident 32KB)
        v2f bfrag;
        bfrag.x = W[(size_t)ka * C64 + c0 + lm];
        bfrag.y = W[(size_t)(ka + 1) * C64 + c0 + lm];

        // D = A x B + C   (8 args: neg_a, A, neg_b, B, c_mod, C, reuse_a, reuse_b)
        acc = __builtin_amdgcn_wmma_f32_16x16x4_f32(
                  false, a, false, bfrag, (short)0, acc, false, false);
    }

    // ---- epilogue: bias + relu, C/D layout: VGPR v -> M = v + half*8, N = lm
    const float bcol = bias[c0 + lm];
    #pragma unroll
    for (int v = 0; v < 8; ++v) {
        int m = v + half * 8;
        float val = acc[v] + bcol;
        val = val > 0.f ? val : 0.f;
        out[(size_t)(row0 + r0 + m) * C64 + c0 + lm] = val;
    }
}

// ---------------------------------------------------------------------------
// Launch
// ---------------------------------------------------------------------------
extern "C" void kernel_launch(void* const* d_in, const int* in_sizes, int n_in,
                              void* d_out, int out_size, void* d_ws, size_t ws_size,
                              hipStream_t stream) {
    const float* x          = (const float*)d_in[0];  // [N,64]
    const int*   edge_index = (const int*)d_in[1];    // [2,E]
    const float* W          = (const float*)d_in[2];  // [128,64]
    const float* b          = (const float*)d_in[3];  // [64]
    float*       out        = (float*)d_out;          // [N,64]

    const int N = in_sizes[0] / C64;      // 100000
    const int E = in_sizes[1] / 2;        // 1000000

    float* sm = (float*)d_ws;             // [N,64] scatter-max workspace (25.6 MB)

    // 1) init sm = -inf
    const int n4 = (N * C64) / 4;
    k_init<<<(n4 + 255) / 256, 256, 0, stream>>>((float4*)sm, n4);

    // 2) scatter-max x[src] into sm[dst]
    const long long ethreads = (long long)E * 16;
    k_edge<<<(unsigned)((ethreads + 255) / 256), 256, 0, stream>>>(
        x, edge_index, edge_index + E, sm, E);

    // 3) fused concat + GEMM + bias + relu (N divisible by 32 -> full blocks, EXEC all-ones)
    k_gemm<<<N / 32, 256, 0, stream>>>(x, sm, W, b, out, N);
}